// MyTopModel_42477226557450
// MI455X (gfx1250) — compile-verified
//
#include <hip/hip_runtime.h>
#include <math.h>

// Problem constants
#define H   768
#define FF  3072
#define NB  16
#define S   2048
#define T   20

typedef __attribute__((ext_vector_type(2))) float v2f;
typedef __attribute__((ext_vector_type(8))) float v8f;

#define ACT_NONE 0
#define ACT_GELU 1
#define ACT_TANH 2

// ---------------------------------------------------------------------------
// Templated K-loop. A is required to be K-contiguous (sak==1): fragment loads
// are contiguous v2f -> global_load_b64 with immediate offsets. Each wave
// accumulates one or two 16x16 C tiles (stacked in M) sharing one B fragment.
// khalf is pre-folded into pa0/pa1/pb by the caller.
// ---------------------------------------------------------------------------
template <bool TWO, bool SBK1>
__device__ __forceinline__ void wmma_kloop(
    const float* __restrict__ pa0, const float* __restrict__ pa1,
    const float* __restrict__ pb, int K, long long sbk,
    v8f& acc0, v8f& acc1)
{
#pragma unroll 4
  for (int k = 0; k < K; k += 4) {
    v2f a0 = *(const v2f*)(pa0 + k);
    v2f a1;
    if (TWO) a1 = *(const v2f*)(pa1 + k);
    v2f b;
    if (SBK1) {
      b = *(const v2f*)(pb + k);
    } else {
      const float* p = pb + (long long)k * sbk;
      b.x = p[0];
      b.y = p[sbk];
    }
    acc0 = __builtin_amdgcn_wmma_f32_16x16x4_f32(
        false, a0, false, b, (short)0, acc0, false, false);
    if (TWO)
      acc1 = __builtin_amdgcn_wmma_f32_16x16x4_f32(
          false, a1, false, b, (short)0, acc1, false, false);
  }
}

// ---------------------------------------------------------------------------
// Strided, batched GEMM on f32 WMMA (V_WMMA_F32_16X16X4_F32).
// C[m,n] = act( alpha * sum_k A[m,k]*B[k,n] + bias_n[n] + bias_m[m] )
// A elem (m,k) at A[m*sam + k]        (A must be K-contiguous)
// B elem (k,n) at B[k*sbk + n*sbn].
// blockDim = 256 (8 waves along N); each wave owns a 32(M) x 16(N) slab
// (two stacked 16x16 WMMA tiles sharing the B fragment).
// Block tile = 32 (M) x 128 (N). grid.z = batch.
// ---------------------------------------------------------------------------
__global__ __launch_bounds__(256) void gemm_wmma_f32(
    const float* __restrict__ A, const float* __restrict__ B,
    float* __restrict__ C,
    int M, int N, int K,
    long long sam, long long sbk, long long sbn, long long scm,
    long long batA, long long batB, long long batC,
    const float* __restrict__ bias_n, const float* __restrict__ bias_m,
    float alpha, int act)
{
  const int bz = blockIdx.z;
  A += (long long)bz * batA;
  B += (long long)bz * batB;
  C += (long long)bz * batC;

  const int wave = threadIdx.x >> 5;     // 0..7 (all along N)
  const int lane = threadIdx.x & 31;
  const int row0 = blockIdx.y * 32;
  const int col0 = blockIdx.x * 128 + wave * 16;
  if (col0 >= N) return;                 // wave-uniform exit: EXEC stays all-1s

  const int lidx  = lane & 15;
  const int khalf = (lane >> 4) << 1;    // 0 or 2 : K half owned by this lane

  int m0 = row0 + lidx;      if (m0 > M - 1) m0 = M - 1;   // clamp; OOB rows never stored
  int m1 = row0 + 16 + lidx; if (m1 > M - 1) m1 = M - 1;
  int nf = col0 + lidx;      if (nf > N - 1) nf = N - 1;

  const float* __restrict__ pa0 = A + (long long)m0 * sam + khalf;
  const float* __restrict__ pa1 = A + (long long)m1 * sam + khalf;
  const float* __restrict__ pb  = B + (long long)nf * sbn + (long long)khalf * sbk;
  const bool two = (row0 + 16 < M);      // wave-uniform

  v8f acc0 = {}, acc1 = {};
  if (two) {
    if (sbk == 1) wmma_kloop<true,  true >(pa0, pa1, pb, K, sbk, acc0, acc1);
    else          wmma_kloop<true,  false>(pa0, pa1, pb, K, sbk, acc0, acc1);
  } else {
    if (sbk == 1) wmma_kloop<false, true >(pa0, pa1, pb, K, sbk, acc0, acc1);
    else          wmma_kloop<false, false>(pa0, pa1, pb, K, sbk, acc0, acc1);
  }

  // ---- epilogue ----
  const int ncol = col0 + lidx;
  const bool nok = (ncol < N);
  const float bn = (bias_n && nok) ? bias_n[ncol] : 0.0f;
  const int half8 = (lane >> 4) << 3;

#pragma unroll
  for (int t = 0; t < 2; ++t) {
    if (t == 1 && !two) break;
    const v8f& acc = (t == 0) ? acc0 : acc1;
    const int mbase = row0 + t * 16 + half8;
#pragma unroll
    for (int i = 0; i < 8; ++i) {
      int mrow = mbase + i;
      if (mrow >= M || !nok) continue;
      float v = acc[i] * alpha + bn;
      if (bias_m) v += bias_m[mrow];
      if (act == ACT_GELU)      v = 0.5f * v * (1.0f + erff(v * 0.70710678118654752f));
      else if (act == ACT_TANH) v = tanhf(v);
      C[(long long)mrow * scm + ncol] = v;
    }
  }
}

// ---------------------------------------------------------------------------
// Block-wide sum over 256 threads (wave32).
// ---------------------------------------------------------------------------
__device__ __forceinline__ float block_sum(float v, float* red) {
  __syncthreads();
  for (int off = 16; off; off >>= 1) v += __shfl_xor(v, off, 32);
  const int wave = threadIdx.x >> 5;
  const int lane = threadIdx.x & 31;
  if (lane == 0) red[wave] = v;
  __syncthreads();
  if (threadIdx.x == 0) {
    float s = 0.0f;
#pragma unroll
    for (int i = 0; i < 8; ++i) s += red[i];
    red[0] = s;
  }
  __syncthreads();
  return red[0];
}

// ---------------------------------------------------------------------------
// out[row] = LN( X[row] + c1*R[row % rmod] + auxS[row]*auxV ) * g + b
// One 256-thread block per row of width 768.
// ---------------------------------------------------------------------------
__global__ __launch_bounds__(256) void ln768_kernel(
    const float* __restrict__ X, const float* __restrict__ R,
    long long rstride, int rmod,
    const float* __restrict__ auxS, const float* __restrict__ auxV,
    const float* __restrict__ g, const float* __restrict__ b,
    float* __restrict__ out, float c1)
{
  __shared__ float red[8];
  __shared__ float buf[H];
  const int row = blockIdx.x;
  const float* Xr = X + (long long)row * H;
  const float* Rr = R ? (R + (long long)(row % rmod) * rstride) : nullptr;
  const float s = auxS ? auxS[row] : 0.0f;

  float part = 0.0f;
  for (int i = threadIdx.x; i < H; i += 256) {
    float v = Xr[i];
    if (Rr)   v += c1 * Rr[i];
    if (auxV) v += s * auxV[i];
    buf[i] = v;
    part += v;
  }
  const float mean = block_sum(part, red) * (1.0f / (float)H);
  float vp = 0.0f;
  for (int i = threadIdx.x; i < H; i += 256) {
    float d = buf[i] - mean;
    vp += d * d;
  }
  const float var  = block_sum(vp, red) * (1.0f / (float)H);
  const float rstd = rsqrtf(var + 1e-5f);
  float* Or = out + (long long)row * H;
  for (int i = threadIdx.x; i < H; i += 256)
    Or[i] = (buf[i] - mean) * rstd * g[i] + b[i];
}

// ---------------------------------------------------------------------------
// rs[row] = sum_c X[row, c]   (one wave per row)
// ---------------------------------------------------------------------------
__global__ __launch_bounds__(256) void rowsum_kernel(
    const float* __restrict__ X, float* __restrict__ rs, int rows, int cols)
{
  const int row = blockIdx.x * 8 + (threadIdx.x >> 5);
  const int lane = threadIdx.x & 31;
  if (row >= rows) return;
  const float* Xr = X + (long long)row * cols;
  float v = 0.0f;
  for (int c = lane; c < cols; c += 32) v += Xr[c];
  for (int off = 16; off; off >>= 1) v += __shfl_xor(v, off, 32);
  if (lane == 0) rs[row] = v;
}

// ---------------------------------------------------------------------------
// out[i*m + j] = f( scale * dot(X[i,:], Y[j,:]) ), one wave per (i,j).
// sigm != 0 -> sigmoid.
// ---------------------------------------------------------------------------
__global__ __launch_bounds__(256) void dots_kernel(
    const float* __restrict__ X, const float* __restrict__ Y,
    float* __restrict__ out, int n, int m, int K, float scale, int sigm)
{
  const int idx = blockIdx.x * 8 + (threadIdx.x >> 5);
  const int lane = threadIdx.x & 31;
  if (idx >= n * m) return;
  const int i = idx / m, j = idx % m;
  const float* Xr = X + (long long)i * K;
  const float* Yr = Y + (long long)j * K;
  float v = 0.0f;
  for (int k = lane; k < K; k += 32) v += Xr[k] * Yr[k];
  for (int off = 16; off; off >>= 1) v += __shfl_xor(v, off, 32);
  if (lane == 0) {
    float r = v * scale;
    if (sigm) r = 1.0f / (1.0f + expf(-r));
    out[idx] = r;
  }
}

// ---------------------------------------------------------------------------
// Branch-2 tiny attention: per batch b (one block):
//   sc[t] = tanh( dot(qpx[b], kt[t]) / sqrt(H) )
//   att2[b,h] = sum_t sc[t] * vt[t,h]
// ---------------------------------------------------------------------------
__global__ __launch_bounds__(256) void attn2_kernel(
    const float* __restrict__ qpx, const float* __restrict__ kt,
    const float* __restrict__ vt, float* __restrict__ att2, float inv)
{
  __shared__ float sc[T];
  const int b = blockIdx.x;
  const int wave = threadIdx.x >> 5;
  const int lane = threadIdx.x & 31;
  const float* qr = qpx + (long long)b * H;
  for (int t = wave; t < T; t += 8) {
    const float* kr = kt + (long long)t * H;
    float v = 0.0f;
    for (int k = lane; k < H; k += 32) v += qr[k] * kr[k];
    for (int off = 16; off; off >>= 1) v += __shfl_xor(v, off, 32);
    if (lane == 0) sc[t] = tanhf(v * inv);
  }
  __syncthreads();
  for (int h = threadIdx.x; h < H; h += 256) {
    float a = 0.0f;
#pragma unroll
    for (int t = 0; t < T; ++t) a += sc[t] * vt[(long long)t * H + h];
    att2[(long long)b * H + h] = a;
  }
}

// ---------------------------------------------------------------------------
// Host-side launch helpers
// ---------------------------------------------------------------------------
static inline void launch_gemm(hipStream_t st,
    const float* A, const float* B, float* C,
    int M, int N, int K,
    long long sam, long long sbk, long long sbn, long long scm,
    long long bA, long long bB, long long bC, int nb,
    const float* bias_n, const float* bias_m, float alpha, int act)
{
  dim3 grid((N + 127) / 128, (M + 31) / 32, nb);
  gemm_wmma_f32<<<grid, 256, 0, st>>>(A, B, C, M, N, K,
      sam, sbk, sbn, scm, bA, bB, bC, bias_n, bias_m, alpha, act);
}

static inline void launch_ln(hipStream_t st,
    const float* X, const float* R, long long rstride, int rmod,
    const float* auxS, const float* auxV,
    const float* g, const float* b, float* out, int rows, float c1)
{
  ln768_kernel<<<dim3(rows), 256, 0, st>>>(X, R, rstride, rmod, auxS, auxV, g, b, out, c1);
}

extern "C" void kernel_launch(void* const* d_in, const int* in_sizes, int n_in,
                              void* d_out, int out_size, void* d_ws, size_t ws_size,
                              hipStream_t stream)
{
  (void)in_sizes; (void)n_in; (void)out_size; (void)ws_size;
  // setup_inputs() dict order (recursive insertion order)
  const float* hidden  = (const float*)d_in[0];   // (16,2048,768)
  const float* Wq      = (const float*)d_in[1];   // (768,768)
  const float* bq      = (const float*)d_in[2];
  const float* Wk      = (const float*)d_in[3];
  const float* bk      = (const float*)d_in[4];
  const float* Wv      = (const float*)d_in[5];
  const float* bv      = (const float*)d_in[6];
  const float* attg    = (const float*)d_in[7];
  const float* attb    = (const float*)d_in[8];
  const float* f1W1    = (const float*)d_in[9];   // (768,3072)
  const float* f1b1    = (const float*)d_in[10];
  const float* f1W2    = (const float*)d_in[11];  // (3072,768)
  const float* f1b2    = (const float*)d_in[12];
  const float* l1g     = (const float*)d_in[13];  // ffn1_ln
  const float* l1b     = (const float*)d_in[14];
  const float* f2W1    = (const float*)d_in[15];
  const float* f2b1    = (const float*)d_in[16];
  const float* f2W2    = (const float*)d_in[17];
  const float* f2b2    = (const float*)d_in[18];
  const float* l2g     = (const float*)d_in[19];  // ffn2_ln
  const float* l2b     = (const float*)d_in[20];
  const float* targets = (const float*)d_in[21];  // (20,768)

  float* out1 = (float*)d_out;          // (16,20,768)
  float* out2 = out1 + NB * T * H;      // (16,20)

  // workspace carve-up
  float* w = (float*)d_ws;
  auto alloc = [&](size_t n) { float* p = w; w += n; return p; };
  float* h1t   = alloc((size_t)T * FF);        // 20x3072
  float* q1t   = alloc((size_t)T * H);
  float* qt    = alloc((size_t)T * H);
  float* qpt   = alloc((size_t)T * H);
  float* qkt   = alloc((size_t)T * H);
  float* sbias = alloc(32);                    // 20 (scaled q.bk)
  float* tn    = alloc((size_t)NB * T * S);    // 16x20x2048
  float* rs    = alloc((size_t)NB * T);        // 320
  float* Amat  = alloc((size_t)NB * T * H);    // 320x768
  float* att1  = alloc((size_t)NB * T * H);
  float* z1    = alloc((size_t)NB * T * H);
  float* h2a   = alloc((size_t)NB * T * FF);   // 320x3072
  float* o1    = alloc((size_t)NB * T * H);
  float* h1x   = alloc((size_t)NB * FF);       // 16x3072
  float* q1x   = alloc((size_t)NB * H);
  float* qx    = alloc((size_t)NB * H);
  float* qpx   = alloc((size_t)NB * H);
  float* ktb   = alloc((size_t)T * H);
  float* vtb   = alloc((size_t)T * H);
  float* att2  = alloc((size_t)NB * H);
  float* z2    = alloc((size_t)NB * H);
  float* h2x   = alloc((size_t)NB * FF);
  float* o2    = alloc((size_t)NB * H);
  float* y2    = alloc((size_t)NB * H);

  const float inv = 1.0f / sqrtf((float)H);
  const long long hb = (long long)S * H;       // hidden batch stride

  // ===== Branch 1: query path on targets (batch-independent) =====
  launch_gemm(stream, targets, f1W1, h1t, T, FF, H, H, FF,1, FF, 0,0,0,1, f1b1, nullptr, 1.0f, ACT_GELU);
  launch_gemm(stream, h1t, f1W2, q1t, T, H, FF, FF, H,1, H, 0,0,0,1, f1b2, nullptr, 1.0f, ACT_NONE);
  launch_ln(stream, q1t, targets, H, T, nullptr, nullptr, l1g, l1b, qt, T, 0.5f);
  launch_gemm(stream, qt, Wq, qpt, T, H, H, H, H,1, H, 0,0,0,1, bq, nullptr, 1.0f, ACT_NONE);
  // qkt = qpt @ Wk^T  (B elem (k,n) = Wk[n*H + k] -> sbk=1, sbn=H)
  launch_gemm(stream, qpt, Wk, qkt, T, H, H, H, 1,H, H, 0,0,0,1, nullptr, nullptr, 1.0f, ACT_NONE);
  // sbias[t] = dot(qpt[t], bk) / sqrt(H)
  dots_kernel<<<dim3((T + 7) / 8), 256, 0, stream>>>(qpt, bk, sbias, T, 1, H, inv, 0);

  // ===== Branch 1: heavy batched passes over hidden_states =====
  // tn[b] = tanh( qkt @ hidden[b]^T * inv + sbias )   (B elem (k,n) = hidden[b][n*H+k] -> sbk=1)
  launch_gemm(stream, qkt, hidden, tn, T, S, H, H, 1,H, S, 0, hb, (long long)T * S, NB,
              nullptr, sbias, inv, ACT_TANH);
  rowsum_kernel<<<dim3((NB * T + 7) / 8), 256, 0, stream>>>(tn, rs, NB * T, S);
  // Amat[b] = tn[b] @ hidden[b]
  launch_gemm(stream, tn, hidden, Amat, T, H, S, S, H,1, H, (long long)T * S, hb, (long long)T * H, NB,
              nullptr, nullptr, 1.0f, ACT_NONE);
  // att1 = Amat @ Wv   (the rs[m]*bv term is folded into the LN below)
  launch_gemm(stream, Amat, Wv, att1, NB * T, H, H, H, H,1, H, 0,0,0,1, nullptr, nullptr, 1.0f, ACT_NONE);
  // z1 = LN(att1 + rs*bv + 0.5*qpt[t])
  launch_ln(stream, att1, qpt, H, T, rs, bv, attg, attb, z1, NB * T, 0.5f);
  // ffn2 + final LN -> out1
  launch_gemm(stream, z1, f2W1, h2a, NB * T, FF, H, H, FF,1, FF, 0,0,0,1, f2b1, nullptr, 1.0f, ACT_GELU);
  launch_gemm(stream, h2a, f2W2, o1, NB * T, H, FF, FF, H,1, H, 0,0,0,1, f2b2, nullptr, 1.0f, ACT_NONE);
  launch_ln(stream, o1, z1, H, NB * T, nullptr, nullptr, l2g, l2b, out1, NB * T, 0.5f);

  // ===== Branch 2: only row 0 of each batch matters =====
  // x[b] = hidden[b,0,:]  -> A strided with sam = S*H
  launch_gemm(stream, hidden, f1W1, h1x, NB, FF, H, hb, FF,1, FF, 0,0,0,1, f1b1, nullptr, 1.0f, ACT_GELU);
  launch_gemm(stream, h1x, f1W2, q1x, NB, H, FF, FF, H,1, H, 0,0,0,1, f1b2, nullptr, 1.0f, ACT_NONE);
  launch_ln(stream, q1x, hidden, hb, NB, nullptr, nullptr, l1g, l1b, qx, NB, 0.5f);
  launch_gemm(stream, qx, Wq, qpx, NB, H, H, H, H,1, H, 0,0,0,1, bq, nullptr, 1.0f, ACT_NONE);
  launch_gemm(stream, targets, Wk, ktb, T, H, H, H, H,1, H, 0,0,0,1, bk, nullptr, 1.0f, ACT_NONE);
  launch_gemm(stream, targets, Wv, vtb, T, H, H, H, H,1, H, 0,0,0,1, bv, nullptr, 1.0f, ACT_NONE);
  attn2_kernel<<<dim3(NB), 256, 0, stream>>>(qpx, ktb, vtb, att2, inv);
  launch_ln(stream, att2, qpx, H, NB, nullptr, nullptr, attg, attb, z2, NB, 0.5f);
  launch_gemm(stream, z2, f2W1, h2x, NB, FF, H, H, FF,1, FF, 0,0,0,1, f2b1, nullptr, 1.0f, ACT_GELU);
  launch_gemm(stream, h2x, f2W2, o2, NB, H, FF, FF, H,1, H, 0,0,0,1, f2b2, nullptr, 1.0f, ACT_NONE);
  launch_ln(stream, o2, z2, H, NB, nullptr, nullptr, l2g, l2b, y2, NB, 0.5f);
  // out2[b,t] = sigmoid( y2[b] . targets[t] )
  dots_kernel<<<dim3((NB * T + 7) / 8), 256, 0, stream>>>(y2, targets, out2, NB, T, H, 1.0f, 1);
}